// PointNet_SA_Module_64381559767676
// MI455X (gfx1250) — compile-verified
//
#include <hip/hip_runtime.h>
#include <cstdint>
#include <cstddef>

// ---- problem constants (from reference) ----
#define RADIUS   0.2f
#define K_NB     32
#define S_OUT    1024
#define BN_EPS   1e-5f
#define B_SZ     8
#define N_PTS    8192
#define D_IN     64
#define M_ROWS   (B_SZ * S_OUT * K_NB)   // 262144 rows for the GEMMs

typedef float v2f __attribute__((ext_vector_type(2)));
typedef float v8f __attribute__((ext_vector_type(8)));

// ---- CDNA5 async-tensor path: global -> LDS without touching VGPR data ----
// LDS address is taken from the vdst VGPR; global address from the 64-bit vaddr VGPRs (no SGPR base).
// Low 32 bits of a generic pointer to __shared__ are the LDS byte offset.
__device__ __forceinline__ void async_load_f32_to_lds(const float* gptr, void* lds_ptr) {
  const uint32_t loff = (uint32_t)(uintptr_t)lds_ptr;
  const uint64_t ga   = (uint64_t)(uintptr_t)gptr;
  asm volatile("global_load_async_to_lds_b32 %0, %1, off"
               :: "v"(loff), "v"(ga) : "memory");
}
__device__ __forceinline__ void wait_asynccnt0() {
  asm volatile("s_wait_asynccnt 0x0" ::: "memory");
}

// =====================================================================
// 1) Farthest-point sampling: one block per batch, 1024 threads,
//    8 points cached in registers per thread; wave32 shuffle argmax.
//    Also writes new_xyz (B,3,S) directly into d_out.
// =====================================================================
__global__ __launch_bounds__(1024) void fps_kernel(const float* __restrict__ xyz,
                                                   int* __restrict__ idx_ws,
                                                   float* __restrict__ out_newxyz) {
  const int b    = blockIdx.x;
  const int t    = threadIdx.x;
  const int lane = t & 31;
  const int wid  = t >> 5;
  const float* X = xyz + (size_t)b * 3 * N_PTS;
  const float* Y = X + N_PTS;
  const float* Z = Y + N_PTS;

  __shared__ float swv[32];
  __shared__ int   swi[32];
  __shared__ int   s_last;
  __shared__ int   s_hist[S_OUT];

  float px[8], py[8], pz[8], dist[8];
#pragma unroll
  for (int i = 0; i < 8; ++i) {
    const int n = t + i * 1024;
    px[i] = X[n]; py[i] = Y[n]; pz[i] = Z[n];
    dist[i] = 3.4e38f;
  }

  int cur = 0;
  for (int s = 0; s < S_OUT; ++s) {
    if (t == 0) s_hist[s] = cur;
    const float cx = X[cur], cy = Y[cur], cz = Z[cur];   // uniform broadcast load
    float bv = -1.f; int bi = 0;
#pragma unroll
    for (int i = 0; i < 8; ++i) {
      const float dx = px[i] - cx, dy = py[i] - cy, dz = pz[i] - cz;
      const float d2 = dx * dx + dy * dy + dz * dz;
      const float nd = fminf(dist[i], d2);
      dist[i] = nd;
      if (nd > bv) { bv = nd; bi = t + i * 1024; }
    }
    // wave32 argmax reduce (prefer smaller index on ties, like jnp.argmax)
#pragma unroll
    for (int off = 16; off > 0; off >>= 1) {
      const float ov = __shfl_xor(bv, off);
      const int   oi = __shfl_xor(bi, off);
      if (ov > bv || (ov == bv && oi < bi)) { bv = ov; bi = oi; }
    }
    if (lane == 0) { swv[wid] = bv; swi[wid] = bi; }
    __syncthreads();
    if (wid == 0) {
      bv = swv[lane]; bi = swi[lane];
#pragma unroll
      for (int off = 16; off > 0; off >>= 1) {
        const float ov = __shfl_xor(bv, off);
        const int   oi = __shfl_xor(bi, off);
        if (ov > bv || (ov == bv && oi < bi)) { bv = ov; bi = oi; }
      }
      if (lane == 0) s_last = bi;
    }
    __syncthreads();
    cur = s_last;
  }

  if (t < S_OUT) {
    const int j = s_hist[t];
    idx_ws[b * S_OUT + t] = j;
    out_newxyz[((size_t)b * 3 + 0) * S_OUT + t] = X[j];
    out_newxyz[((size_t)b * 3 + 1) * S_OUT + t] = Y[j];
    out_newxyz[((size_t)b * 3 + 2) * S_OUT + t] = Z[j];
  }
}

// =====================================================================
// 2) Ball query: one wave32 per center; ballot + prefix-popc compaction
//    preserves ascending index order == reference's sort()[:K].
// =====================================================================
__global__ __launch_bounds__(256) void query_ball_kernel(const float* __restrict__ xyz,
                                                         const int* __restrict__ idx_ws,
                                                         int* __restrict__ gidx) {
  const int lane = threadIdx.x & 31;
  const int wave = blockIdx.x * (blockDim.x >> 5) + (threadIdx.x >> 5); // center id
  const int b = wave / S_OUT;
  const int s = wave - b * S_OUT;
  const float* X = xyz + (size_t)b * 3 * N_PTS;
  const float* Y = X + N_PTS;
  const float* Z = Y + N_PTS;
  const int j0 = idx_ws[b * S_OUT + s];
  const float cx = X[j0], cy = Y[j0], cz = Z[j0];
  int* out = gidx + (size_t)wave * K_NB;
  const float r2 = RADIUS * RADIUS;

  int cnt = 0;
  int first = N_PTS - 1;                       // matches min(idx, N-1) when empty
  for (int n0 = 0; n0 < N_PTS && cnt < K_NB; n0 += 32) {
    const int n = n0 + lane;
    const float dx = X[n] - cx, dy = Y[n] - cy, dz = Z[n] - cz;
    const bool pred = (dx * dx + dy * dy + dz * dz) <= r2;
    const unsigned mask = (unsigned)__ballot(pred);   // wave32: low 32 bits
    if (cnt == 0 && mask) first = n0 + __ffs(mask) - 1;
    const int pos = __popc(mask & ((1u << lane) - 1u));
    if (pred && (cnt + pos) < K_NB) out[cnt + pos] = n;
    cnt += __popc(mask);
  }
  for (int kk = cnt + lane; kk < K_NB; kk += 32) out[kk] = first;
}

// =====================================================================
// 3) Transpose points (B,64,N) -> (B,N,64) so gather rows are contiguous
// =====================================================================
__global__ __launch_bounds__(256) void transpose_kernel(const float* __restrict__ in,
                                                        float* __restrict__ out) {
  __shared__ float tile[32][33];
  const int b = blockIdx.z;
  const int nbase = blockIdx.x * 32;
  const int dbase = blockIdx.y * 32;
  const int tx = threadIdx.x;   // 0..31
  const int ty = threadIdx.y;   // 0..7
  const float* src = in + ((size_t)b * D_IN + dbase) * N_PTS + nbase;
#pragma unroll
  for (int i = 0; i < 4; ++i)
    tile[ty + i * 8][tx] = src[(size_t)(ty + i * 8) * N_PTS + tx];
  __syncthreads();
  float* dst = out + ((size_t)b * N_PTS + nbase) * D_IN + dbase;
#pragma unroll
  for (int i = 0; i < 4; ++i)
    dst[(size_t)(ty + i * 8) * D_IN + tx] = tile[tx][ty + i * 8];
}

// =====================================================================
// 4) Gather + concat -> feat (M x 68): [points(64) | (xyz-center)/R (3) | 0 pad]
//    16 threads per row, float4 bursts (256B per row).
// =====================================================================
__global__ __launch_bounds__(256) void gather_kernel(const float* __restrict__ xyz,
                                                     const float* __restrict__ pts_n,
                                                     const int* __restrict__ idx_ws,
                                                     const int* __restrict__ gidx,
                                                     float* __restrict__ feat) {
  const int tid = threadIdx.x;
  const int rlocal = tid >> 4;
  const int q = tid & 15;
  const int row = blockIdx.x * 16 + rlocal;
  const int b = row / (S_OUT * K_NB);
  const int rem = row - b * (S_OUT * K_NB);
  const int s = rem >> 5;          // K_NB == 32
  const int j = gidx[row];
  const float4* srcp = (const float4*)(pts_n + ((size_t)b * N_PTS + j) * D_IN);
  float4* dstp = (float4*)(feat + (size_t)row * 68);
  dstp[q] = srcp[q];
  if (q == 0) {
    const float* X = xyz + (size_t)b * 3 * N_PTS;
    const int j0 = idx_ws[b * S_OUT + s];
    const float inv = 1.0f / RADIUS;
    float* f = feat + (size_t)row * 68;
    f[64] = (X[j] - X[j0]) * inv;
    f[65] = (X[N_PTS + j] - X[N_PTS + j0]) * inv;
    f[66] = (X[2 * N_PTS + j] - X[2 * N_PTS + j0]) * inv;
    f[67] = 0.f;
  }
}

// =====================================================================
// 5) WMMA fp32 GEMM (Y = relu(bn(A)) * W^T + b) + per-channel stats.
//    One wave computes a 16xNOUT strip via V_WMMA_F32_16X16X4_F32.
//    W is async-copied global->LDS (ASYNCcnt path) and transposed on the
//    fly to k-major so B-fragment reads are stride-1 across lanes
//    (bank-conflict-free without padding).
//    FUSE: apply previous layer's BN(scale,shift)+ReLU on the A-load.
// =====================================================================
template <int KIN_REAL, int KIN, int NOUT, bool FUSE>
__global__ __launch_bounds__(256) void gemm_wmma_kernel(const float* __restrict__ A,
                                                        const float* __restrict__ W,
                                                        const float* __restrict__ bias,
                                                        const float* __restrict__ scsh_prev,
                                                        float* __restrict__ Y,
                                                        float* __restrict__ stats) {
  constexpr int NT = NOUT / 16;
  __shared__ float wT[KIN * NOUT];          // k-major: wT[k*NOUT + n] = W[n][k]
  __shared__ float bias_lds[NOUT];
  __shared__ float sc_lds[KIN];
  __shared__ float sh_lds[KIN];
  __shared__ float stats_lds[2 * NOUT];

  const int tid = threadIdx.x;
  // async global->LDS copy of W with on-the-fly transpose (per-lane dsaddr)
  for (int i = tid; i < NOUT * KIN_REAL; i += 256) {
    const int o = i / KIN_REAL, k = i - o * KIN_REAL;
    async_load_f32_to_lds(W + i, &wT[k * NOUT + o]);
  }
  // zero-fill padded K rows with plain ds stores
  for (int i = tid; i < (KIN - KIN_REAL) * NOUT; i += 256)
    wT[KIN_REAL * NOUT + i] = 0.f;
  for (int i = tid; i < NOUT; i += 256) bias_lds[i] = bias[i];
  if constexpr (FUSE) {
    for (int i = tid; i < KIN; i += 256) {
      sc_lds[i] = scsh_prev[i];
      sh_lds[i] = scsh_prev[KIN + i];
    }
  }
  for (int i = tid; i < 2 * NOUT; i += 256) stats_lds[i] = 0.f;
  wait_asynccnt0();                         // s_wait_asynccnt 0
  __syncthreads();

  const int lane = tid & 31, wid = tid >> 5;
  const int l15 = lane & 15, half = lane >> 4;
  const int mbase = (blockIdx.x * 8 + wid) * 16;
  const float* arow = A + (size_t)(mbase + l15) * KIN;
  const int koff = half * 2;

  v8f acc[NT] = {};
  for (int k = 0; k < KIN; k += 4) {
    __builtin_prefetch(arow + k + 32, 0, 1);            // global_prefetch_b8
    float x0 = arow[k + koff];
    float x1 = arow[k + koff + 1];
    if constexpr (FUSE) {
      x0 = fmaxf(0.f, fmaf(x0, sc_lds[k + koff],     sh_lds[k + koff]));
      x1 = fmaxf(0.f, fmaf(x1, sc_lds[k + koff + 1], sh_lds[k + koff + 1]));
    }
    v2f a; a.x = x0; a.y = x1;   // A 16x4: lane<16 holds k,k+1 ; lane>=16 holds k+2,k+3
#pragma unroll
    for (int nt = 0; nt < NT; ++nt) {
      const float* wb = &wT[(k + koff) * NOUT + nt * 16 + l15];
      v2f bb; bb.x = wb[0]; bb.y = wb[NOUT];            // B 4x16 mirrored layout
      acc[nt] = __builtin_amdgcn_wmma_f32_16x16x4_f32(
          /*neg_a=*/false, a, /*neg_b=*/false, bb,
          /*c_mod=*/(short)0, acc[nt], /*reuse_a=*/false, /*reuse_b=*/false);
    }
  }

#pragma unroll
  for (int nt = 0; nt < NT; ++nt) {
    const int col = nt * 16 + l15;
    const float bsv = bias_lds[col];
    float s1 = 0.f, s2 = 0.f;
#pragma unroll
    for (int v = 0; v < 8; ++v) {                       // C: lane<16 M=v ; lane>=16 M=v+8
      const float val = acc[nt][v] + bsv;
      Y[(size_t)(mbase + half * 8 + v) * NOUT + col] = val;
      s1 += val; s2 += val * val;
    }
    s1 += __shfl_xor(s1, 16);                           // fold lane L with L+16 (same col)
    s2 += __shfl_xor(s2, 16);
    if (half == 0) {
      atomicAdd(&stats_lds[col], s1);                   // ds_add_f32
      atomicAdd(&stats_lds[NOUT + col], s2);
    }
  }
  __syncthreads();
  for (int i = tid; i < 2 * NOUT; i += 256)
    unsafeAtomicAdd(&stats[i], stats_lds[i]);           // global_atomic_add_f32
}

// =====================================================================
// 6) Fold BN stats into (scale, shift) per channel
// =====================================================================
__global__ void bn_finalize_kernel(const float* __restrict__ stats,
                                   const float* __restrict__ g,
                                   const float* __restrict__ be,
                                   float* __restrict__ scsh, int nout) {
  const int o = threadIdx.x;
  if (o < nout) {
    const float inv_m = 1.0f / (float)M_ROWS;
    const float mu  = stats[o] * inv_m;
    const float var = stats[nout + o] * inv_m - mu * mu;
    const float sc  = g[o] * rsqrtf(var + BN_EPS);
    scsh[o] = sc;
    scsh[nout + o] = be[o] - mu * sc;
  }
}

// =====================================================================
// 7) Apply final BN+ReLU and max-pool over K; write (B,128,S)
// =====================================================================
__global__ __launch_bounds__(128) void maxpool_out_kernel(const float* __restrict__ y2,
                                                          const float* __restrict__ scsh,
                                                          float* __restrict__ out_pts) {
  const int bs = blockIdx.x;       // b*S + s
  const int o  = threadIdx.x;      // channel 0..127
  const int b = bs / S_OUT, s = bs - b * S_OUT;
  const float sc = scsh[o], sh = scsh[128 + o];
  const float* base = y2 + (size_t)bs * K_NB * 128 + o;
  float m = -3.4e38f;
#pragma unroll 4
  for (int kk = 0; kk < K_NB; ++kk) {
    const float v = fmaxf(0.f, fmaf(base[(size_t)kk * 128], sc, sh));
    m = fmaxf(m, v);
  }
  out_pts[((size_t)b * 128 + o) * S_OUT + s] = m;
}

// =====================================================================
// Launch
// =====================================================================
extern "C" void kernel_launch(void* const* d_in, const int* in_sizes, int n_in,
                              void* d_out, int out_size, void* d_ws, size_t ws_size,
                              hipStream_t stream) {
  (void)in_sizes; (void)n_in; (void)out_size; (void)ws_size;
  const float* xyz    = (const float*)d_in[0];
  const float* points = (const float*)d_in[1];
  const float* W0  = (const float*)d_in[2];
  const float* b0  = (const float*)d_in[3];
  const float* g0  = (const float*)d_in[4];
  const float* be0 = (const float*)d_in[5];
  const float* W1  = (const float*)d_in[6];
  const float* b1  = (const float*)d_in[7];
  const float* g1  = (const float*)d_in[8];
  const float* be1 = (const float*)d_in[9];
  const float* W2  = (const float*)d_in[10];
  const float* b2  = (const float*)d_in[11];
  const float* g2  = (const float*)d_in[12];
  const float* be2 = (const float*)d_in[13];

  float* out_newxyz = (float*)d_out;                             // (B,3,S)
  float* out_pts    = (float*)d_out + (size_t)B_SZ * 3 * S_OUT;  // (B,128,S)

  // workspace carve-up (256B aligned)
  char* ws = (char*)d_ws;
  size_t off = 0;
  auto carve = [&](size_t bytes) {
    void* p = ws + off;
    off += (bytes + 255) & ~(size_t)255;
    return p;
  };
  int*   idx_ws = (int*)  carve((size_t)B_SZ * S_OUT * sizeof(int));
  int*   gidx   = (int*)  carve((size_t)M_ROWS * sizeof(int));
  float* stats0 = (float*)carve(2 * 64  * sizeof(float));
  float* stats1 = (float*)carve(2 * 64  * sizeof(float));
  float* stats2 = (float*)carve(2 * 128 * sizeof(float));
  float* scsh0  = (float*)carve(2 * 64  * sizeof(float));
  float* scsh1  = (float*)carve(2 * 64  * sizeof(float));
  float* scsh2  = (float*)carve(2 * 128 * sizeof(float));
  float* pts_n  = (float*)carve((size_t)B_SZ * N_PTS * D_IN * sizeof(float)); // ~16.8MB
  float* feat   = (float*)carve((size_t)M_ROWS * 68  * sizeof(float));        // ~71MB
  float* y0     = (float*)carve((size_t)M_ROWS * 64  * sizeof(float));        // ~67MB
  float* y1     = (float*)carve((size_t)M_ROWS * 64  * sizeof(float));        // ~67MB
  float* y2     = (float*)carve((size_t)M_ROWS * 128 * sizeof(float));        // ~134MB

  hipMemsetAsync(stats0, 0, 2 * 64  * sizeof(float), stream);
  hipMemsetAsync(stats1, 0, 2 * 64  * sizeof(float), stream);
  hipMemsetAsync(stats2, 0, 2 * 128 * sizeof(float), stream);

  fps_kernel<<<B_SZ, 1024, 0, stream>>>(xyz, idx_ws, out_newxyz);

  query_ball_kernel<<<(B_SZ * S_OUT) / 8, 256, 0, stream>>>(xyz, idx_ws, gidx);

  transpose_kernel<<<dim3(N_PTS / 32, D_IN / 32, B_SZ), dim3(32, 8), 0, stream>>>(points, pts_n);

  gather_kernel<<<M_ROWS / 16, 256, 0, stream>>>(xyz, pts_n, idx_ws, gidx, feat);

  // layer 0: feat(M x 68pad) -> y0(M x 64)
  gemm_wmma_kernel<67, 68, 64, false><<<M_ROWS / 128, 256, 0, stream>>>(
      feat, W0, b0, nullptr, y0, stats0);
  bn_finalize_kernel<<<1, 64, 0, stream>>>(stats0, g0, be0, scsh0, 64);

  // layer 1: relu(bn(y0)) -> y1(M x 64)
  gemm_wmma_kernel<64, 64, 64, true><<<M_ROWS / 128, 256, 0, stream>>>(
      y0, W1, b1, scsh0, y1, stats1);
  bn_finalize_kernel<<<1, 64, 0, stream>>>(stats1, g1, be1, scsh1, 64);

  // layer 2: relu(bn(y1)) -> y2(M x 128)
  gemm_wmma_kernel<64, 64, 128, true><<<M_ROWS / 128, 256, 0, stream>>>(
      y1, W2, b2, scsh1, y2, stats2);
  bn_finalize_kernel<<<1, 128, 0, stream>>>(stats2, g2, be2, scsh2, 128);

  // final BN+ReLU + max over K, transpose to (B,128,S)
  maxpool_out_kernel<<<B_SZ * S_OUT, 128, 0, stream>>>(y2, scsh2, out_pts);
}